// Group_13048110645536
// MI455X (gfx1250) — compile-verified
//
#include <hip/hip_runtime.h>

typedef __attribute__((ext_vector_type(2))) float v2f;
typedef __attribute__((ext_vector_type(8))) float v8f;
typedef __attribute__((ext_vector_type(4))) unsigned int v4u;
typedef __attribute__((ext_vector_type(8))) unsigned int v8u;

#define NPTS 8192
#define NCTR 512
#define NNBR 32
#define NBATCH 16

// ---------------------------------------------------------------------------
// Kernel 1: farthest point sampling, one workgroup per batch.
// Points + running min-dist live in LDS; 511 sequential argmax iterations.
// ---------------------------------------------------------------------------
extern "C" __global__ __launch_bounds__(256)
void fps_kernel(const float* __restrict__ x, float* __restrict__ centers)
{
    extern __shared__ __align__(16) float smem[];
    float* px   = smem;                // [NPTS]
    float* py   = px + NPTS;           // [NPTS]
    float* pz   = py + NPTS;           // [NPTS]
    float* mind = pz + NPTS;           // [NPTS]
    float* rv   = mind + NPTS;         // [8]  per-wave argmax value
    int*   ri   = (int*)(rv + 8);      // [8]  per-wave argmax index
    int*   curp = ri + 8;              // [1]  broadcast of selected index

    const int b   = blockIdx.x;
    const int tid = threadIdx.x;
    const float* xb = x + (size_t)b * NPTS * 3;

    for (int i = tid; i < NPTS; i += 256) {
        px[i]   = xb[i * 3 + 0];
        py[i]   = xb[i * 3 + 1];
        pz[i]   = xb[i * 3 + 2];
        mind[i] = 1e10f;
    }
    if (tid == 0) {
        float* c0 = centers + (size_t)b * NCTR * 3;
        c0[0] = xb[0]; c0[1] = xb[1]; c0[2] = xb[2];
        curp[0] = 0;
    }
    __syncthreads();

    int last = 0;
    for (int k = 1; k < NCTR; ++k) {
        const float lx = px[last], ly = py[last], lz = pz[last];
        float bv = -1.0f;
        int   bi = 0x7fffffff;
        for (int i = tid; i < NPTS; i += 256) {
            float dx = px[i] - lx, dy = py[i] - ly, dz = pz[i] - lz;
            float d  = dx * dx + dy * dy + dz * dz;
            float m  = fminf(mind[i], d);
            mind[i]  = m;
            if (m > bv) { bv = m; bi = i; }   // i ascending -> lowest idx on tie
        }
        // wave32 argmax reduction, tie -> lowest index (matches jnp.argmax)
        #pragma unroll
        for (int mask = 16; mask >= 1; mask >>= 1) {
            float ov = __shfl_xor(bv, mask, 32);
            int   oi = __shfl_xor(bi, mask, 32);
            if (ov > bv || (ov == bv && oi < bi)) { bv = ov; bi = oi; }
        }
        if ((tid & 31) == 0) { rv[tid >> 5] = bv; ri[tid >> 5] = bi; }
        __syncthreads();
        if (tid == 0) {
            float fv = rv[0]; int fi = ri[0];
            #pragma unroll
            for (int w = 1; w < 8; ++w)
                if (rv[w] > fv || (rv[w] == fv && ri[w] < fi)) { fv = rv[w]; fi = ri[w]; }
            curp[0] = fi;
            float* c = centers + ((size_t)b * NCTR + k) * 3;
            c[0] = px[fi]; c[1] = py[fi]; c[2] = pz[fi];
        }
        __syncthreads();
        last = curp[0];
    }
}

// Monotonic key transform: total order on float bits matching float compare.
__device__ __forceinline__ unsigned int f32_key(float f) {
    unsigned int u = __float_as_uint(f);
    return u ^ ((unsigned int)((int)u >> 31) | 0x80000000u);
}

// ---------------------------------------------------------------------------
// Kernel 2: KNN(32) + gather + recenter.
// Points staged LDS-resident via the Tensor Data Mover (1-D D# per wave),
// distances via V_WMMA_F32_16X16X4_F32:
//   A row m = (-2cx, -2cy, -2cz, |c|^2), B col n = (x, y, z, 1)
//   d2 = C[m][n] + |x_n|^2
// Neighbors extracted ascending by packed u64 (d2,idx) threshold passes.
// ---------------------------------------------------------------------------
extern "C" __global__ __launch_bounds__(128)
void knn_kernel(const float* __restrict__ x, const float* __restrict__ centers,
                float* __restrict__ groups)
{
    extern __shared__ __align__(16) float smem2[];
    float4* pts = (float4*)smem2;              // [NPTS] (x, y, z, |p|^2)
    float*  raw = smem2 + NPTS * 4;            // [NPTS*3] TDM landing zone
    float*  cen = raw + NPTS * 3;              // [4 waves][16][3]

    const int b        = blockIdx.x >> 3;      // 8 tiles (of 64 centers) per batch
    const int tileBase = (blockIdx.x & 7) * 64;
    const int tid  = threadIdx.x;
    const int lane = tid & 31;
    const int wave = tid >> 5;
    const int half = lane >> 4;                // 0: lanes 0-15, 1: lanes 16-31
    const int l16  = lane & 15;

    const float* xb = x + (size_t)b * NPTS * 3;

    // --- TDM: each wave DMAs one quarter (6144 f32 = 24576 B) of the cloud ---
    {
        const unsigned long long gaddr =
            (unsigned long long)(xb) + (unsigned long long)wave * 24576ull;
        const unsigned int ldsoff =
            (unsigned int)(unsigned long long)(raw) + (unsigned int)wave * 24576u;
        v4u g0;
        g0.x = 1u;                                     // count=1, user mode
        g0.y = ldsoff;                                 // lds_addr (bytes)
        g0.z = (unsigned int)gaddr;                    // global_addr[31:0]
        g0.w = ((unsigned int)(gaddr >> 32) & 0x01FFFFFFu) | (2u << 30); // [56:32] | type=2
        v8u g1;
        g1.s0 = 2u << 16;          // data_size=4B, workgroup_mask=0
        g1.s1 = 6144u << 16;       // tensor_dim0[15:0] in [31:16]
        g1.s2 = (0u) | (1u << 16); // tensor_dim0[31:16]=0, tensor_dim1[15:0]=1
        g1.s3 = 6144u << 16;       // tensor_dim1 hi=0, tile_dim0=6144
        g1.s4 = 1u;                // tile_dim1=1, tile_dim2=0
        g1.s5 = 6144u;             // tensor_dim0_stride[31:0]
        g1.s6 = 0u;                // stride0 hi, tensor_dim1_stride lo
        g1.s7 = 0u;
        asm volatile("tensor_load_to_lds %0, %1"
                     :
                     : "s"(g0), "s"(g1)
                     : "memory");
        __builtin_amdgcn_s_wait_tensorcnt(0);
    }
    __syncthreads();

    // Expand raw xyz -> float4 (x, y, z, |p|^2)
    for (int i = tid; i < NPTS; i += 128) {
        float ax = raw[i * 3 + 0], ay = raw[i * 3 + 1], az = raw[i * 3 + 2];
        pts[i] = make_float4(ax, ay, az, ax * ax + ay * ay + az * az);
    }

    const int cBase = tileBase + wave * 16;    // this wave's first center row
    const float* cb = centers + ((size_t)b * NCTR + cBase) * 3;
    const float cx = cb[l16 * 3 + 0];
    const float cy = cb[l16 * 3 + 1];
    const float cz = cb[l16 * 3 + 2];
    const float csq = cx * cx + cy * cy + cz * cz;
    if (lane < 16) {
        cen[(wave * 16 + lane) * 3 + 0] = cx;
        cen[(wave * 16 + lane) * 3 + 1] = cy;
        cen[(wave * 16 + lane) * 3 + 2] = cz;
    }
    __syncthreads();

    // A-matrix per ISA layout: vgpr0 holds K={0,2}, vgpr1 holds K={1,3},
    // split across the two 16-lane halves.
    v2f amat;
    amat.x = half ? (-2.0f * cz) : (-2.0f * cx);
    amat.y = half ? csq          : (-2.0f * cy);

    unsigned long long prevk[8];
    #pragma unroll
    for (int r = 0; r < 8; ++r) prevk[r] = 0ull;

    float* gout = groups + ((size_t)b * NCTR + cBase) * NNBR * 3;

    for (int nb = 0; nb < NNBR; ++nb) {
        unsigned long long bestk[8];
        #pragma unroll
        for (int r = 0; r < 8; ++r) bestk[r] = ~0ull;

        for (int j = 0; j < NPTS; j += 16) {
            const float4 p = pts[j + l16];
            v2f bmat;
            bmat.x = half ? p.z  : p.x;
            bmat.y = half ? 1.0f : p.y;
            v8f c = {};
            c = __builtin_amdgcn_wmma_f32_16x16x4_f32(
                    false, amat, false, bmat, (short)0, c, false, false);
            const unsigned int idx = (unsigned int)(j + l16);
            #pragma unroll
            for (int r = 0; r < 8; ++r) {
                // C layout: vgpr r -> row M=r (lanes 0-15) / M=8+r (lanes 16-31)
                const unsigned long long key =
                    ((unsigned long long)f32_key(c[r] + p.w) << 32) | idx;
                if (key > prevk[r] && key < bestk[r]) bestk[r] = key;
            }
        }
        // u64 min-reduce within each 16-lane half; all lanes get the result
        #pragma unroll
        for (int mask = 1; mask <= 8; mask <<= 1) {
            #pragma unroll
            for (int r = 0; r < 8; ++r) {
                unsigned long long ok = __shfl_xor(bestk[r], mask, 32);
                if (ok < bestk[r]) bestk[r] = ok;
            }
        }
        #pragma unroll
        for (int r = 0; r < 8; ++r) prevk[r] = bestk[r];

        // one writer lane per half emits its 8 center rows
        if (l16 == 0) {
            #pragma unroll
            for (int r = 0; r < 8; ++r) {
                const int m = r + half * 8;
                const float4 p = pts[(unsigned int)bestk[r] & 0xFFFFu];
                float* o = gout + ((size_t)m * NNBR + nb) * 3;
                o[0] = p.x - cen[(wave * 16 + m) * 3 + 0];
                o[1] = p.y - cen[(wave * 16 + m) * 3 + 1];
                o[2] = p.z - cen[(wave * 16 + m) * 3 + 2];
            }
        }
    }
}

// ---------------------------------------------------------------------------
extern "C" void kernel_launch(void* const* d_in, const int* in_sizes, int n_in,
                              void* d_out, int out_size, void* d_ws, size_t ws_size,
                              hipStream_t stream) {
    (void)in_sizes; (void)n_in; (void)out_size; (void)d_ws; (void)ws_size;
    const float* x = (const float*)d_in[0];
    float* groups  = (float*)d_out;                                    // (16,512,32,3)
    float* centers = (float*)d_out + (size_t)NBATCH * NCTR * NNBR * 3; // (16,512,3)

    const size_t shm1 = (size_t)(4 * NPTS + 8) * sizeof(float)
                      + 9 * sizeof(int);                               // ~128.1 KB
    fps_kernel<<<NBATCH, 256, shm1, stream>>>(x, centers);

    const size_t shm2 = (size_t)(NPTS * 4 + NPTS * 3 + 4 * 16 * 3) * sizeof(float); // ~225 KB
    knn_kernel<<<NBATCH * 8, 128, shm2, stream>>>(x, centers, groups);
}